// Corr2D_15453292331292
// MI455X (gfx1250) — compile-verified
//
#include <hip/hip_runtime.h>
#include <stdint.h>

#define H  6144
#define W  6144
#define OH 6142
#define OW 6142

#define TILE_Y 32
#define TILE_X 64
#define IN_Y   (TILE_Y + 2)   /* 34 */
#define IN_X   (TILE_X + 2)   /* 66 */

typedef float        v2f __attribute__((ext_vector_type(2)));
typedef float        v8f __attribute__((ext_vector_type(8)));
typedef unsigned int v4u __attribute__((ext_vector_type(4)));
typedef unsigned int v8u __attribute__((ext_vector_type(8)));

__global__ __launch_bounds__(256) void conv3x3_tdm_wmma(const float* __restrict__ x,
                                                        const float* __restrict__ kern,
                                                        const float* __restrict__ bias,
                                                        float* __restrict__ out)
{
    __shared__ float tile[IN_Y * IN_X];

    const int row0 = blockIdx.y * TILE_Y;
    const int col0 = blockIdx.x * TILE_X;

    // ---- Stage input halo tile (34 x 66 f32) into LDS with the Tensor Data Mover.
    // One wave issues the DMA; TDM zero-fills reads past tensor_dim (edge tiles).
    if (threadIdx.x < 32) {
        const uint64_t gaddr  = (uint64_t)(uintptr_t)x +
                                ((uint64_t)row0 * (uint64_t)W + (uint64_t)col0) * 4ull;
        const uint32_t ldsOff = (uint32_t)(uintptr_t)(&tile[0]); // low 32b of flat LDS ptr = LDS offset

        v4u g0;                                        // D# group 0 (128b)
        g0.x = 1u;                                     // count=1 (valid), user descriptor
        g0.y = ldsOff;                                 // lds_addr (bytes)
        g0.z = (uint32_t)(gaddr & 0xffffffffull);      // global_addr[31:0]
        g0.w = (uint32_t)(gaddr >> 32) | (2u << 30);   // global_addr[56:32] | type=2 ("image")

        v8u g1;                                        // D# group 1 (256b)
        g1[0] = (2u << 16);                            // workgroup_mask=0 | data_size=2 (4B)
        g1[1] = ((uint32_t)W & 0xffffu) << 16;         // tensor_dim0[15:0] in [31:16]
        g1[2] = ((uint32_t)H & 0xffffu) << 16;         // tensor_dim0[31:16]=0 | tensor_dim1[15:0]
        g1[3] = ((uint32_t)IN_X) << 16;                // tensor_dim1[31:16]=0 | tile_dim0=66
        g1[4] = (uint32_t)IN_Y;                        // tile_dim1=34 | tile_dim2=0
        g1[5] = (uint32_t)W;                           // tensor_dim0_stride[31:0]=6144
        g1[6] = 0u;                                    // stride hi | tensor_dim1_stride lo
        g1[7] = 0u;                                    // tensor_dim1_stride hi

        asm volatile("tensor_load_to_lds %0, %1" :: "s"(g0), "s"(g1) : "memory");
        __builtin_amdgcn_s_wait_tensorcnt(0);
    }
    __syncthreads();

    // ---- Per-wave 16x16 output tile: banded-matmul conv on the WMMA pipe.
    const int lane = threadIdx.x & 31;
    const int wv   = threadIdx.x >> 5;        // 8 waves: 2 (y) x 4 (x) tiles
    const int oy   = (wv >> 2) * 16;
    const int ox   = (wv & 3)  * 16;
    const int half = lane >> 4;               // K-half for A/B operands
    const int n    = lane & 15;               // A: M row index; B/C/D: N column index

#if defined(__has_builtin) && __has_builtin(__builtin_amdgcn_wmma_f32_16x16x4_f32)
    v8f acc = {0.f, 0.f, 0.f, 0.f, 0.f, 0.f, 0.f, 0.f};

    #pragma unroll
    for (int i = 0; i < 3; ++i) {             // vertical tap
        const float kr0 = kern[3*i + 0];
        const float kr1 = kern[3*i + 1];
        const float kr2 = kern[3*i + 2];
        const int   arow = (oy + n + i) * IN_X;
        #pragma unroll
        for (int c = 0; c < 5; ++c) {         // K=18 band split into 5 chunks of 4
            const int pb = 4*c + 2*half;
            // A[m,p] = X[oy+m+i, ox+p]; clamp cols into tile (their B entries are 0.0)
            int c0 = ox + pb;     if (c0 > IN_X - 1) c0 = IN_X - 1;
            int c1 = ox + pb + 1; if (c1 > IN_X - 1) c1 = IN_X - 1;
            v2f a; a.x = tile[arow + c0]; a.y = tile[arow + c1];
            // B[p,n] = K[i, p-n] for p-n in [0,2], else 0 (banded)
            const int d0 = pb - n;
            const int d1 = d0 + 1;
            v2f b;
            b.x = (d0 == 0) ? kr0 : ((d0 == 1) ? kr1 : ((d0 == 2) ? kr2 : 0.f));
            b.y = (d1 == 0) ? kr0 : ((d1 == 1) ? kr1 : ((d1 == 2) ? kr2 : 0.f));
            acc = __builtin_amdgcn_wmma_f32_16x16x4_f32(
                      false, a, false, b, (short)0, acc, false, false);
        }
    }

    // C/D layout: VGPR r holds M=r (lanes 0-15) and M=r+8 (lanes 16-31), N=lane%16
    const float bv  = bias[0];
    const int   gx  = col0 + ox + n;
    const int   gy0 = row0 + oy + half * 8;
    if (gx < OW) {
        #pragma unroll
        for (int r = 0; r < 8; ++r) {
            const int gy = gy0 + r;
            if (gy < OH) out[(size_t)gy * OW + gx] = acc[r] + bv;
        }
    }
#else
    // VALU fallback (kept only so a builtin-signature mismatch can't kill the TDM path)
    const int lx = threadIdx.x & 63;
    const int ly = (threadIdx.x >> 6) * 8;
    const float bv = bias[0];
    for (int r = 0; r < 8; ++r) {
        const int yy = ly + r;
        float s = 0.f;
        #pragma unroll
        for (int i = 0; i < 3; ++i)
            #pragma unroll
            for (int j = 0; j < 3; ++j)
                s += kern[3*i + j] * tile[(yy + i) * IN_X + lx + j];
        const int gy = row0 + yy, gx2 = col0 + lx;
        if (gy < OH && gx2 < OW) out[(size_t)gy * OW + gx2] = s + bv;
    }
#endif
}

extern "C" void kernel_launch(void* const* d_in, const int* in_sizes, int n_in,
                              void* d_out, int out_size, void* d_ws, size_t ws_size,
                              hipStream_t stream) {
    (void)in_sizes; (void)n_in; (void)out_size; (void)d_ws; (void)ws_size;
    const float* x    = (const float*)d_in[0];
    const float* kern = (const float*)d_in[1];
    const float* bias = (const float*)d_in[2];
    float*       out  = (float*)d_out;

    dim3 grid((OW + TILE_X - 1) / TILE_X,   // 96
              (OH + TILE_Y - 1) / TILE_Y);  // 192
    conv3x3_tdm_wmma<<<grid, 256, 0, stream>>>(x, kern, bias, out);
}